// GCNClusterNet_20014547599875
// MI455X (gfx1250) — compile-verified
//
#include <hip/hip_runtime.h>
#include <hip/hip_bf16.h>

typedef __attribute__((ext_vector_type(2))) float v2f;
typedef __attribute__((ext_vector_type(8))) float v8f;

#define N_NODES   100000
#define N_EDGES   3200000
#define D_IN      512
#define D_HID     512
#define D_EMB     256
#define K_CL      64
#define CL_TEMP   30.0f
#define NUM_ITER  10
#define BPAD      4          // LDS row pad: keeps 16B alignment + kills bank conflicts

// ---------------------------------------------------------------------------
// f32 WMMA GEMM with LDS-staged B strip: C[M,N] = A[M,K] @ Bt[N,K]^T.
// Block = 8 waves, one 64-wide N strip (Bs: 64 x K <= 128KB of the WGP's
// 320KB LDS, loaded cooperatively once). Each wave does a 16x64 C strip with
// 4 accumulators. Hot loop: 1 pipelined global_load_b64 (A) + 4 ds_load_b64.
// Early-exit for ragged M happens after the only barrier; EXEC all-ones at
// every WMMA.
// ---------------------------------------------------------------------------
__global__ void wmma_gemm_lds(const float* __restrict__ A,
                              const float* __restrict__ Bt,
                              float* __restrict__ C,
                              int M, int N, int K) {
    __shared__ float Bs[64 * (512 + BPAD)];        // sized for K <= 512
    const int ldb = K + BPAD;
    const int tN  = blockIdx.y << 6;

    // cooperative fill: 64 rows x K floats, float4 granularity
    int rowQuads = K >> 2;
    int total    = rowQuads << 6;                  // 64 * K/4
    const float4* Bg = (const float4*)(Bt + (size_t)tN * K);
    for (int idx = threadIdx.x; idx < total; idx += 256) {
        int row = idx / rowQuads;
        int c4  = idx - row * rowQuads;
        float4 v = Bg[(size_t)row * rowQuads + c4];
        *(float4*)(Bs + row * ldb + (c4 << 2)) = v;
    }
    __syncthreads();

    int wave = blockIdx.x * 8 + (threadIdx.x >> 5);
    int tM   = wave << 4;
    if (tM >= M) return;                           // wave-uniform, after barrier

    int lane = threadIdx.x & 31;
    int half = lane >> 4;                          // 0: K pair {0,1}; 1: {2,3}
    int l    = lane & 15;
    int ko   = half << 1;

    const float* Arow = A + (size_t)(tM + l) * K + ko;
    const float* L0   = Bs + ( 0 + l) * ldb + ko;
    const float* L1   = Bs + (16 + l) * ldb + ko;
    const float* L2   = Bs + (32 + l) * ldb + ko;
    const float* L3   = Bs + (48 + l) * ldb + ko;

    v8f acc0 = {}, acc1 = {}, acc2 = {}, acc3 = {};

    v2f a = *(const v2f*)(Arow);                   // prologue A frag (k = 0)
    int k = 0;
    #pragma unroll 2
    for (; k < K - 4; k += 4) {
        v2f an = *(const v2f*)(Arow + k + 4);      // next A frag in flight
        v2f b0 = *(const v2f*)(L0 + k);            // ds_load_b64, conflict-free
        v2f b1 = *(const v2f*)(L1 + k);
        v2f b2 = *(const v2f*)(L2 + k);
        v2f b3 = *(const v2f*)(L3 + k);
        acc0 = __builtin_amdgcn_wmma_f32_16x16x4_f32(false, a, false, b0, (short)0, acc0, false, false);
        acc1 = __builtin_amdgcn_wmma_f32_16x16x4_f32(false, a, false, b1, (short)0, acc1, false, false);
        acc2 = __builtin_amdgcn_wmma_f32_16x16x4_f32(false, a, false, b2, (short)0, acc2, false, false);
        acc3 = __builtin_amdgcn_wmma_f32_16x16x4_f32(false, a, false, b3, (short)0, acc3, false, false);
        a = an;
    }
    {   // epilogue: last K-step
        v2f b0 = *(const v2f*)(L0 + k);
        v2f b1 = *(const v2f*)(L1 + k);
        v2f b2 = *(const v2f*)(L2 + k);
        v2f b3 = *(const v2f*)(L3 + k);
        acc0 = __builtin_amdgcn_wmma_f32_16x16x4_f32(false, a, false, b0, (short)0, acc0, false, false);
        acc1 = __builtin_amdgcn_wmma_f32_16x16x4_f32(false, a, false, b1, (short)0, acc1, false, false);
        acc2 = __builtin_amdgcn_wmma_f32_16x16x4_f32(false, a, false, b2, (short)0, acc2, false, false);
        acc3 = __builtin_amdgcn_wmma_f32_16x16x4_f32(false, a, false, b3, (short)0, acc3, false, false);
    }

    // C layout: VGPR i -> M = i (lanes 0-15) / M = 8+i (lanes 16-31)
    float* Crow = C + (size_t)(tM + (half << 3)) * N + tN + l;
    #pragma unroll
    for (int i = 0; i < 8; ++i) {
        Crow[(size_t)i * N +  0] = acc0[i];
        Crow[(size_t)i * N + 16] = acc1[i];
        Crow[(size_t)i * N + 32] = acc2[i];
        Crow[(size_t)i * N + 48] = acc3[i];
    }
}

// ---------------------------------------------------------------------------
// cluster_mean: C[64,256] += R^T @ D, split-K with atomic merge,
// software-pipelined b32 fragment loads.
// ---------------------------------------------------------------------------
__global__ void wmma_atb_splitk(const float* __restrict__ R,
                                const float* __restrict__ D,
                                float* __restrict__ C,
                                int Krows, int kChunk) {
    int wave = (blockIdx.x * blockDim.x + threadIdx.x) >> 5;
    int lane = threadIdx.x & 31;
    const int tilesM = K_CL / 16, tilesN = D_EMB / 16;   // 4 x 16 = 64 tiles
    int tileId = wave % (tilesM * tilesN);
    int split  = wave / (tilesM * tilesN);
    int tM = (tileId / tilesN) << 4;
    int tN = (tileId % tilesN) << 4;
    int k0 = split * kChunk;
    int k1 = k0 + kChunk; if (k1 > Krows) k1 = Krows;

    int half = lane >> 4;
    int l    = lane & 15;

    const float* Rp = R + (size_t)(k0 + (half << 1)) * K_CL  + tM + l;
    const float* Dp = D + (size_t)(k0 + (half << 1)) * D_EMB + tN + l;

    v8f acc = {};
    v2f a, b;
    a.x = Rp[0];       a.y = Rp[K_CL];
    b.x = Dp[0];       b.y = Dp[D_EMB];
    Rp += (size_t)4 * K_CL;  Dp += (size_t)4 * D_EMB;

    #pragma unroll 2
    for (int k = k0 + 4; k < k1; k += 4) {
        v2f an, bn;
        an.x = Rp[0];  an.y = Rp[K_CL];
        bn.x = Dp[0];  bn.y = Dp[D_EMB];
        Rp += (size_t)4 * K_CL;  Dp += (size_t)4 * D_EMB;
        acc = __builtin_amdgcn_wmma_f32_16x16x4_f32(false, a, false, b, (short)0, acc, false, false);
        a = an; b = bn;
    }
    acc = __builtin_amdgcn_wmma_f32_16x16x4_f32(false, a, false, b, (short)0, acc, false, false);

    #pragma unroll
    for (int i = 0; i < 8; ++i) {
        int row = tM + (half << 3) + i;
        atomicAdd(&C[row * D_EMB + tN + l], acc[i]);
    }
}

// ---------------------------------------------------------------------------
// CSR-by-destination build: histogram -> block scan -> scatter.
// ---------------------------------------------------------------------------
__global__ void zero_i32(int* __restrict__ p, int n) {
    int i = blockIdx.x * blockDim.x + threadIdx.x;
    if (i < n) p[i] = 0;
}

__global__ void count_dst(const int* __restrict__ dst, int* __restrict__ counts) {
    int e = blockIdx.x * blockDim.x + threadIdx.x;   // grid sized exactly N_EDGES
    atomicAdd(&counts[dst[e]], 1);
}

// In-place per-block inclusive scan (Hillis-Steele in LDS) + block sums.
__global__ void scan_block(int* __restrict__ data, int* __restrict__ bsums, int n) {
    __shared__ int s[256];
    int i = blockIdx.x * 256 + threadIdx.x;
    int v = (i < n) ? data[i] : 0;
    s[threadIdx.x] = v;
    __syncthreads();
    #pragma unroll
    for (int off = 1; off < 256; off <<= 1) {
        int t = (threadIdx.x >= off) ? s[threadIdx.x - off] : 0;
        __syncthreads();
        s[threadIdx.x] += t;
        __syncthreads();
    }
    if (i < n) data[i] = s[threadIdx.x];
    if (threadIdx.x == 255) bsums[blockIdx.x] = s[255];
}

__global__ void scan_sums(int* __restrict__ bsums, int nb) {   // one block, nb <= 512
    __shared__ int s[512];
    int t = threadIdx.x;
    s[t] = (t < nb) ? bsums[t] : 0;
    __syncthreads();
    #pragma unroll
    for (int off = 1; off < 512; off <<= 1) {
        int v = (t >= off) ? s[t - off] : 0;
        __syncthreads();
        s[t] += v;
        __syncthreads();
    }
    if (t < nb) bsums[t] = s[t];                 // inclusive scan of block sums
}

__global__ void scan_finalize(const int* __restrict__ scanned,
                              const int* __restrict__ bsums,
                              int* __restrict__ row_start, int n) {
    int i = blockIdx.x * 256 + threadIdx.x;
    if (i < n) {
        int add = (blockIdx.x > 0) ? bsums[blockIdx.x - 1] : 0;
        row_start[i + 1] = scanned[i] + add;
    }
    if (i == 0) row_start[0] = 0;
}

__global__ void copy_i32(const int* __restrict__ a, int* __restrict__ b, int n) {
    int i = blockIdx.x * blockDim.x + threadIdx.x;
    if (i < n) b[i] = a[i];
}

__global__ void scatter_edges(const int* __restrict__ dst,
                              int* __restrict__ cursor,
                              int* __restrict__ edge_ids) {
    int e = blockIdx.x * blockDim.x + threadIdx.x;   // grid sized exactly N_EDGES
    int p = atomicAdd(&cursor[dst[e]], 1);
    edge_ids[p] = e;
}

// ---------------------------------------------------------------------------
// CSR SPMM (gather form, zero f32 atomics): one wave per destination node.
// hout[node] = bias + sum_{e in row} w[e] * hin[src[e]]   (float4 lanes)
// ---------------------------------------------------------------------------
template <int D>
__global__ void spmm_csr(const int* __restrict__ row_start,
                         const int* __restrict__ edge_ids,
                         const int* __restrict__ src,
                         const float* __restrict__ w,
                         const float* __restrict__ hin,
                         const float* __restrict__ bias,
                         float* __restrict__ hout) {
    constexpr int NV = D / 128;                 // float4 chunks per lane
    int node = (blockIdx.x * blockDim.x + threadIdx.x) >> 5;
    int lane = threadIdx.x & 31;
    if (node >= N_NODES) return;                // wave-uniform

    int beg = row_start[node], end = row_start[node + 1];

    float4 acc[NV];
    const float4* b4 = (const float4*)bias;
    #pragma unroll
    for (int i = 0; i < NV; ++i) acc[i] = b4[lane + (i << 5)];

    for (int t = beg; t < end; ++t) {
        int   e  = edge_ids[t];
        int   s  = src[e];
        float ww = w[e];
        const float4* r4 = (const float4*)(hin + (size_t)s * D);
        #pragma unroll
        for (int i = 0; i < NV; ++i) {
            float4 v = r4[lane + (i << 5)];
            acc[i].x = fmaf(ww, v.x, acc[i].x);
            acc[i].y = fmaf(ww, v.y, acc[i].y);
            acc[i].z = fmaf(ww, v.z, acc[i].z);
            acc[i].w = fmaf(ww, v.w, acc[i].w);
        }
    }
    float4* o4 = (float4*)(hout + (size_t)node * D);
    #pragma unroll
    for (int i = 0; i < NV; ++i) o4[lane + (i << 5)] = acc[i];
}

// ---------------------------------------------------------------------------
// 16x16-tiled LDS transpose: out[cols,rows] = in[rows,cols]^T.
// ---------------------------------------------------------------------------
__global__ void transpose_f32(const float* __restrict__ in, float* __restrict__ out,
                              int rows, int cols) {
    __shared__ float tile[16][17];
    int x = blockIdx.x * 16 + threadIdx.x;
    int y = blockIdx.y * 16 + threadIdx.y;
    if (x < cols && y < rows) tile[threadIdx.y][threadIdx.x] = in[y * cols + x];
    __syncthreads();
    int tx = blockIdx.y * 16 + threadIdx.x;     // index into rows
    int ty = blockIdx.x * 16 + threadIdx.y;     // index into cols
    if (tx < rows && ty < cols) out[ty * rows + tx] = tile[threadIdx.x][threadIdx.y];
}

__global__ void zero_f32(float* __restrict__ p, int n) {
    int i = blockIdx.x * blockDim.x + threadIdx.x;
    if (i < n) p[i] = 0.0f;
}

// Row L2-normalize: data[n] = e[n]/||e[n]||, one wave per 256-wide row.
__global__ void row_normalize(const float* __restrict__ e,
                              float* __restrict__ data, int n) {
    int wave = (blockIdx.x * blockDim.x + threadIdx.x) >> 5;
    int lane = threadIdx.x & 31;
    if (wave >= n) return;
    const float* row = e + (size_t)wave * D_EMB;
    float ss = 0.0f;
    #pragma unroll
    for (int i = 0; i < D_EMB / 32; ++i) {
        float v = row[lane + 32 * i];
        ss += v * v;
    }
    #pragma unroll
    for (int off = 16; off > 0; off >>= 1)
        ss += __shfl_xor(ss, off, 32);
    float inv = rsqrtf(ss);
    #pragma unroll
    for (int i = 0; i < D_EMB / 32; ++i)
        data[(size_t)wave * D_EMB + lane + 32 * i] = row[lane + 32 * i] * inv;
}

// Row softmax over 64 clusters (2 vals/lane, wave32) + optional cluster_r
// via LDS staging then 64 global atomics per block.
__global__ void softmax_rows(const float* __restrict__ dist,
                             float* __restrict__ r,
                             float* __restrict__ cluster_r, int n) {
    __shared__ float sc[K_CL];
    if (cluster_r && threadIdx.x < K_CL) sc[threadIdx.x] = 0.0f;
    __syncthreads();

    int wave = (blockIdx.x * blockDim.x + threadIdx.x) >> 5;
    int lane = threadIdx.x & 31;
    if (wave < n) {
        const float* dr = dist + (size_t)wave * K_CL;
        float v0 = CL_TEMP * dr[lane];
        float v1 = CL_TEMP * dr[lane + 32];
        float m = fmaxf(v0, v1);
        #pragma unroll
        for (int off = 16; off > 0; off >>= 1)
            m = fmaxf(m, __shfl_xor(m, off, 32));
        float e0 = __expf(v0 - m), e1 = __expf(v1 - m);
        float s = e0 + e1;
        #pragma unroll
        for (int off = 16; off > 0; off >>= 1)
            s += __shfl_xor(s, off, 32);
        float inv = 1.0f / s;
        e0 *= inv; e1 *= inv;
        float* rr = r + (size_t)wave * K_CL;
        rr[lane]      = e0;
        rr[lane + 32] = e1;
        if (cluster_r) {
            atomicAdd(&sc[lane], e0);            // ds_add_f32
            atomicAdd(&sc[lane + 32], e1);
        }
    }
    __syncthreads();
    if (cluster_r && threadIdx.x < K_CL)
        atomicAdd(&cluster_r[threadIdx.x], sc[threadIdx.x]);
}

__global__ void mu_update(const float* __restrict__ cm,
                          const float* __restrict__ cr,
                          float* __restrict__ mu) {
    int i = blockIdx.x * blockDim.x + threadIdx.x;
    if (i < K_CL * D_EMB) mu[i] = cm[i] / cr[i / D_EMB];
}

// ---------------------------------------------------------------------------
extern "C" void kernel_launch(void* const* d_in, const int* in_sizes, int n_in,
                              void* d_out, int out_size, void* d_ws, size_t ws_size,
                              hipStream_t stream) {
    const float* x       = (const float*)d_in[0];
    const int*   esrc    = (const int*)  d_in[1];
    const int*   edst    = (const int*)  d_in[2];
    const float* ew      = (const float*)d_in[3];
    const float* W1      = (const float*)d_in[4];
    const float* b1      = (const float*)d_in[5];
    const float* W2      = (const float*)d_in[6];
    const float* b2      = (const float*)d_in[7];
    const float* mu_init = (const float*)d_in[8];
    // d_in[9] = num_iter (device scalar) -> deterministic 10, hardcoded
    (void)in_sizes; (void)n_in; (void)out_size; (void)ws_size;

    // Output: mu[64,256] | r[100000,64] | embeds[100000,256] | dist[100000,64]
    float* out      = (float*)d_out;
    float* out_mu   = out;
    float* out_r    = out + K_CL * D_EMB;
    float* out_emb  = out_r + (size_t)N_NODES * K_CL;
    float* out_dist = out_emb + (size_t)N_NODES * D_EMB;

    // Workspace layout (8B alignment preserved for all float2-loaded regions):
    float* ws   = (float*)d_ws;
    float* buf0 = ws;                                    // 51.2M f: h0 -> h2
    float* buf1 = ws + (size_t)N_NODES * D_HID;          // 51.2M f: h  -> data
    int*   iw        = (int*)(buf1 + (size_t)N_NODES * D_HID);
    int*   counts    = iw;                               // 100000 (scan in-place)
    int*   row_start = iw + 100000;                      // 100002 (padded)
    int*   cursor    = iw + 200002;                      // 100000
    int*   bsums     = iw + 300002;                      // 512
    int*   edge_ids  = iw + 300514;                      // 3,200,000  (total ints even)
    float* fw  = (float*)(edge_ids + N_EDGES);           // 8B-aligned
    float* W1t = fw;                                     // 512*512
    float* W2t = fw + D_IN * D_HID;                      // 256*512
    float* mu  = W2t + D_EMB * D_HID;                    // 64*256
    float* cm  = mu + K_CL * D_EMB;                      // 64*256
    float* cr  = cm + K_CL * D_EMB;                      // 64 (contiguous after cm)

    const int nScanBlk = (N_NODES + 255) / 256;          // 391
    const int nMBlk    = (N_NODES / 16 + 7) / 8;         // 782 M-blocks (8 waves each)

    // --- CSR-by-destination build (int atomics only) ---
    zero_i32<<<nScanBlk, 256, 0, stream>>>(counts, N_NODES);
    count_dst<<<N_EDGES / 256, 256, 0, stream>>>(edst, counts);
    scan_block<<<nScanBlk, 256, 0, stream>>>(counts, bsums, N_NODES);
    scan_sums<<<1, 512, 0, stream>>>(bsums, nScanBlk);
    scan_finalize<<<nScanBlk, 256, 0, stream>>>(counts, bsums, row_start, N_NODES);
    copy_i32<<<nScanBlk, 256, 0, stream>>>(row_start, cursor, N_NODES);
    scatter_edges<<<N_EDGES / 256, 256, 0, stream>>>(edst, cursor, edge_ids);

    // --- Pre-transpose weights so all GEMMs take the contiguous-K path ---
    transpose_f32<<<dim3(D_HID / 16, D_IN / 16), dim3(16, 16), 0, stream>>>(W1, W1t, D_IN, D_HID);
    transpose_f32<<<dim3(D_EMB / 16, D_IN / 16), dim3(16, 16), 0, stream>>>(W2, W2t, D_IN, D_EMB);

    // 1) h0 = x @ W1 -> buf0
    wmma_gemm_lds<<<dim3(nMBlk, D_HID / 64), 256, 0, stream>>>(x, W1t, buf0,
                                                               N_NODES, D_HID, D_IN);
    // 2) h = spmm(h0) + b1 -> buf1  (gather CSR, no f32 atomics)
    spmm_csr<D_HID><<<N_NODES / 8, 256, 0, stream>>>(row_start, edge_ids, esrc, ew,
                                                     buf0, b1, buf1);
    // 3) h2 = h @ W2 -> buf0
    wmma_gemm_lds<<<dim3(nMBlk, D_EMB / 64), 256, 0, stream>>>(buf1, W2t, buf0,
                                                               N_NODES, D_EMB, D_HID);
    // 4) embeds = spmm(h2) + b2 -> d_out embeds slot
    spmm_csr<D_EMB><<<N_NODES / 8, 256, 0, stream>>>(row_start, edge_ids, esrc, ew,
                                                     buf0, b2, out_emb);
    // 5) data = row-normalize(embeds) -> buf1
    row_normalize<<<N_NODES / 8, 256, 0, stream>>>(out_emb, buf1, N_NODES);

    // 6) soft k-means, 10 iterations
    hipMemcpyAsync(mu, mu_init, K_CL * D_EMB * sizeof(float),
                   hipMemcpyDeviceToDevice, stream);
    for (int it = 0; it < NUM_ITER; ++it) {
        // dist = data @ mu^T  (mu is [64,256] = [N,K] already)
        wmma_gemm_lds<<<dim3(nMBlk, 1), 256, 0, stream>>>(buf1, mu, out_dist,
                                                          N_NODES, K_CL, D_EMB);
        zero_f32<<<(K_CL * D_EMB + K_CL + 255) / 256, 256, 0, stream>>>(
            cm, K_CL * D_EMB + K_CL);                    // cm and cr together
        softmax_rows<<<N_NODES / 8, 256, 0, stream>>>(out_dist, out_r, cr, N_NODES);
        wmma_atb_splitk<<<1000, 256, 0, stream>>>(out_r, buf1, cm, N_NODES, 800);
        mu_update<<<(K_CL * D_EMB + 255) / 256, 256, 0, stream>>>(cm, cr, mu);
    }

    // 7) final dist / r with converged mu; copy mu out
    wmma_gemm_lds<<<dim3(nMBlk, 1), 256, 0, stream>>>(buf1, mu, out_dist,
                                                      N_NODES, K_CL, D_EMB);
    softmax_rows<<<N_NODES / 8, 256, 0, stream>>>(out_dist, out_r, nullptr, N_NODES);
    hipMemcpyAsync(out_mu, mu, K_CL * D_EMB * sizeof(float),
                   hipMemcpyDeviceToDevice, stream);
}